// ChamferLoss_20134806683972
// MI455X (gfx1250) — compile-verified
//
#include <hip/hip_runtime.h>
#include <hip/hip_bf16.h>

// ---------------------------------------------------------------------------
// Chamfer loss for MI455X (gfx1250), fp32 WMMA path.
//   preds, targ : (64, 32, 4 + 256*128) fp32
//   out         : [action_loss + chamfer_mean, a0_loss]  (2 fp32)
// One workgroup per (b,h) group: x,y (256x128) staged in LDS (320KB/WGP),
// ZZ = x*y^T via v_wmma_f32_16x16x4_f32, fused min/sum epilogue.
// ---------------------------------------------------------------------------

#define OBS_DIM   128
#define NPTS      256
#define T_DIM     (4 + 256 * OBS_DIM)   // 32772
#define NGROUP    2048                  // 64*32
#define LDS_S     132                   // row stride in floats: 132 % 64 == 4 -> conflict-free b64 frag loads

typedef __attribute__((ext_vector_type(2))) float v2f;
typedef __attribute__((ext_vector_type(4))) float v4f;
typedef __attribute__((ext_vector_type(8))) float v8f;

#define FLT_BIG 3.402823466e+38f

// LDS layout (floats)
#define OFF_X    0
#define OFF_Y    (NPTS * LDS_S)                 // 33792
#define OFF_X2   (2 * NPTS * LDS_S)             // 67584
#define OFF_Y2   (OFF_X2 + NPTS)                // 67840
#define OFF_WCM  (OFF_Y2 + NPTS)                // 68096  (8 waves x 256 col-mins)
#define OFF_L2P  (OFF_WCM + 8 * NPTS)           // 70144  (16 loss2 partials)
#define OFF_RED  (OFF_L2P + 16)                 // 70160  (256 tree-reduce)
#define LDS_FLOATS (OFF_RED + NPTS)             // 70416
#define LDS_BYTES  (LDS_FLOATS * 4)             // 281664 <= 320KB

__device__ __forceinline__ v8f wmma_f32_16x16x4(v2f a, v2f b, v8f c) {
    // (neg_a, A, neg_b, B, c_mod, C, reuse_a, reuse_b)
    return __builtin_amdgcn_wmma_f32_16x16x4_f32(false, a, false, b, (short)0, c,
                                                 false, false);
}

__global__ __launch_bounds__(256) void chamfer_group_kernel(
    const float* __restrict__ preds, const float* __restrict__ targ,
    float* __restrict__ ws) {
    extern __shared__ float smem[];
    float* ldsX   = smem + OFF_X;    // x = targ obs (rows n)
    float* ldsY   = smem + OFF_Y;    // y = preds obs (rows m)
    float* ldsX2  = smem + OFF_X2;
    float* ldsY2  = smem + OFF_Y2;
    float* wcm    = smem + OFF_WCM;
    float* l2p    = smem + OFF_L2P;
    float* redbuf = smem + OFF_RED;

    const int g    = blockIdx.x;
    const int tid  = threadIdx.x;
    const int lane = tid & 31;
    const int w    = tid >> 5;        // wave 0..7
    const int hi   = lane >> 4;       // half-wave
    const int li   = lane & 15;

    const size_t base = (size_t)g * T_DIM + 4;
    const float* xg = targ  + base;
    const float* yg = preds + base;

    // ---- Stage 1: global -> LDS (float4, coalesced), padded rows ----
    {
        const v4f* xg4 = reinterpret_cast<const v4f*>(xg);
        const v4f* yg4 = reinterpret_cast<const v4f*>(yg);
        for (int i = tid; i < NPTS * (OBS_DIM / 4); i += 256) {
            int row = i >> 5;
            int c4  = i & 31;
            *reinterpret_cast<v4f*>(&ldsX[row * LDS_S + c4 * 4]) = xg4[i];
            *reinterpret_cast<v4f*>(&ldsY[row * LDS_S + c4 * 4]) = yg4[i];
        }
    }
    __syncthreads();

    // ---- Stage 2: squared norms per point ----
    {
        float sx = 0.0f, sy = 0.0f;
        const float* xr = &ldsX[tid * LDS_S];
        const float* yr = &ldsY[tid * LDS_S];
        #pragma unroll 8
        for (int d = 0; d < OBS_DIM; ++d) {
            sx += xr[d] * xr[d];
            sy += yr[d] * yr[d];
        }
        ldsX2[tid] = sx;
        ldsY2[tid] = sy;
    }
    __syncthreads();

    // ---- Stage 3: per-wave 32-row strip of P = x2 + y2 - 2*x*y^T ----
    const int rbase = w * 32;
    // x2 for the rows this lane's C fragments map to (C: row = r + 8*hi)
    float xs0[8], xs1[8];
    #pragma unroll
    for (int r = 0; r < 8; ++r) {
        xs0[r] = ldsX2[rbase + 8 * hi + r];
        xs1[r] = ldsX2[rbase + 16 + 8 * hi + r];
    }

    float rmin0[8], rmin1[8];
    #pragma unroll
    for (int r = 0; r < 8; ++r) { rmin0[r] = FLT_BIG; rmin1[r] = FLT_BIG; }

    const float* xb0 = &ldsX[(rbase + li) * LDS_S];        // A rows, tile 0
    const float* xb1 = &ldsX[(rbase + 16 + li) * LDS_S];   // A rows, tile 1

    for (int ct = 0; ct < 16; ++ct) {
        const float* yb = &ldsY[(ct * 16 + li) * LDS_S];   // B rows (cols of P)
        v8f c0 = {};
        v8f c1 = {};
        #pragma unroll 8
        for (int ks = 0; ks < 32; ++ks) {
            const int ko = 4 * ks + 2 * hi;   // per-lane K pair within 4-K block
            v2f a0 = *reinterpret_cast<const v2f*>(&xb0[ko]);
            v2f a1 = *reinterpret_cast<const v2f*>(&xb1[ko]);
            v2f b  = *reinterpret_cast<const v2f*>(&yb[ko]);
            c0 = wmma_f32_16x16x4(a0, b, c0);
            c1 = wmma_f32_16x16x4(a1, b, c1);
        }

        // Epilogue: P elements, running row-mins + this tile's column-mins
        const float ycol = ldsY2[ct * 16 + li];
        float cmin = FLT_BIG;
        #pragma unroll
        for (int r = 0; r < 8; ++r) {
            float p0 = xs0[r] + ycol - 2.0f * c0[r];
            float p1 = xs1[r] + ycol - 2.0f * c1[r];
            rmin0[r] = fminf(rmin0[r], p0);
            rmin1[r] = fminf(rmin1[r], p1);
            cmin = fminf(cmin, fminf(p0, p1));
        }
        // combine the two half-waves (same column, rows r vs 8+r)
        cmin = fminf(cmin, __shfl_xor(cmin, 16, 32));
        if (hi == 0) wcm[w * NPTS + ct * 16 + li] = cmin;  // min over this wave's 32 rows
    }

    // Row-min reduce across the 16 lanes of each half (columns)
    #pragma unroll
    for (int mask = 1; mask <= 8; mask <<= 1) {
        #pragma unroll
        for (int r = 0; r < 8; ++r) {
            rmin0[r] = fminf(rmin0[r], __shfl_xor(rmin0[r], mask, 32));
            rmin1[r] = fminf(rmin1[r], __shfl_xor(rmin1[r], mask, 32));
        }
    }
    if (li == 0) {
        float s2 = 0.0f;
        #pragma unroll
        for (int r = 0; r < 8; ++r) s2 += rmin0[r] + rmin1[r];
        l2p[w * 2 + hi] = s2;   // 16 rows' min-sums per half
    }
    __syncthreads();

    // ---- Stage 4: column-min across 8 waves, deterministic tree sums ----
    float cm = wcm[tid];
    #pragma unroll
    for (int ww = 1; ww < 8; ++ww) cm = fminf(cm, wcm[ww * NPTS + tid]);
    redbuf[tid] = cm;
    __syncthreads();
    for (int s = 128; s > 0; s >>= 1) {
        if (tid < s) redbuf[tid] += redbuf[tid + s];
        __syncthreads();
    }

    if (tid == 0) {
        float loss2 = 0.0f;
        #pragma unroll
        for (int i = 0; i < 16; ++i) loss2 += l2p[i];
        float chamfer_g = redbuf[0] + loss2;

        // action MSE for this (b,h)
        const float* pa = preds + (size_t)g * T_DIM;
        const float* ta = targ  + (size_t)g * T_DIM;
        float al = 0.0f;
        #pragma unroll
        for (int i = 0; i < 4; ++i) {
            float d = pa[i] - ta[i];
            al += d * d;
        }
        al *= 0.25f;

        ws[2 * g + 0] = chamfer_g;
        ws[2 * g + 1] = al;
    }
}

__global__ __launch_bounds__(256) void chamfer_finalize_kernel(
    const float* __restrict__ ws, float* __restrict__ out) {
    __shared__ float sc[256], sa[256], s0[256];
    const int t = threadIdx.x;
    float ch = 0.0f, wal = 0.0f, a0 = 0.0f;
    for (int g = t; g < NGROUP; g += 256) {
        float c = ws[2 * g + 0];
        float a = ws[2 * g + 1];
        ch += c;
        bool h0 = ((g & 31) == 0);   // H == 32 -> h==0
        wal += (h0 ? 10.0f : 1.0f) * a;
        if (h0) a0 += a;
    }
    sc[t] = ch; sa[t] = wal; s0[t] = a0;
    __syncthreads();
    for (int s = 128; s > 0; s >>= 1) {
        if (t < s) { sc[t] += sc[t + s]; sa[t] += sa[t + s]; s0[t] += s0[t + s]; }
        __syncthreads();
    }
    if (t == 0) {
        out[0] = sa[0] / (float)NGROUP + sc[0] / (float)NGROUP;
        out[1] = s0[0] / 64.0f;
    }
}

extern "C" void kernel_launch(void* const* d_in, const int* in_sizes, int n_in,
                              void* d_out, int out_size, void* d_ws, size_t ws_size,
                              hipStream_t stream) {
    const float* preds = (const float*)d_in[0];
    const float* targ  = (const float*)d_in[1];
    float* out = (float*)d_out;
    float* ws  = (float*)d_ws;

    // Allow >64KB dynamic LDS (320KB per WGP on CDNA5). Not a stream op; capture-safe.
    (void)hipFuncSetAttribute(reinterpret_cast<const void*>(chamfer_group_kernel),
                              hipFuncAttributeMaxDynamicSharedMemorySize, LDS_BYTES);

    chamfer_group_kernel<<<NGROUP, 256, LDS_BYTES, stream>>>(preds, targ, ws);
    chamfer_finalize_kernel<<<1, 256, 0, stream>>>(ws, out);
}